// SS2D_CAAS_5188320494184
// MI455X (gfx1250) — compile-verified
//
#include <hip/hip_runtime.h>
#include <hip/hip_bf16.h>

// ---------------- problem constants (from reference setup_inputs) ----------
constexpr int Bsz  = 8;
constexpr int Dm   = 96;
constexpr int Himg = 64;
constexpr int Wimg = 64;
constexpr int L    = Himg * Wimg;        // 4096
constexpr int DIN  = 192;
constexpr int KDIR = 4;
constexpr int RLOW = 6;
constexpr int NCH  = 8;                  // R + 2N = 6 + 2
constexpr int HID  = 32;
constexpr int NCHK = 8;                  // scan chunks per sequence
constexpr int CLEN = L / NCHK;           // 512

// ---------------- WMMA types ----------------------------------------------
typedef float v2f __attribute__((ext_vector_type(2)));
typedef float v8f __attribute__((ext_vector_type(8)));

__device__ __forceinline__ v8f wmma_f32_16x16x4(v2f a, v2f b, v8f c) {
  // 8-arg form: (neg_a, A, neg_b, B, c_mod, C, reuse_a, reuse_b)
  return __builtin_amdgcn_wmma_f32_16x16x4_f32(false, a, false, b,
                                               (short)0, c, false, false);
}

__device__ __forceinline__ float silu_f(float x) {
  return x / (1.0f + __expf(-x));
}
__device__ __forceinline__ float sigmoid_f(float x) {
  return 1.0f / (1.0f + __expf(-x));
}
__device__ __forceinline__ float softplus_f(float x) {
  return (x > 20.0f) ? x : log1pf(__expf(x));
}

// CDNA5 async global -> LDS copy of one 16B chunk (per-lane addresses),
// tracked by ASYNCcnt.  ldsaddr = wave-relative LDS byte address (low 32 bits
// of the generic pointer to a __shared__ object, per ISA 10.2 aperture rules).
__device__ __forceinline__ void async_g2l_b128(unsigned ldsaddr, const float* g) {
  asm volatile("global_load_async_to_lds_b128 %0, %1, off"
               :: "v"(ldsaddr), "v"(g) : "memory");
}
__device__ __forceinline__ void wait_asynccnt0() {
  asm volatile("s_wait_asynccnt 0" ::: "memory");
}

// direction map: sequence index l -> spatial index s (h*W+w)
__device__ __forceinline__ int dir_s(int k, int l) {
  switch (k) {
    case 0:  return l;
    case 1:  return ((l & 63) << 6) | (l >> 6);          // l = w*H+h
    case 2:  return L - 1 - l;
    default: { int lp = L - 1 - l; return ((lp & 63) << 6) | (lp >> 6); }
  }
}

// ============================================================
// K1: in_proj GEMM  xz[b, c(384), l] = sum_cin W[c,cin] * x[b,cin,l]
//     Block = (b, mt, group of 8 nt); the 16x96 A tile (contiguous 6KB of W)
//     is staged once into LDS via async-to-LDS and shared by all 8 waves.
// ============================================================
__global__ void k_in_proj(const float* __restrict__ x,
                          const float* __restrict__ w,   // [384, 96]
                          float* __restrict__ xz) {
  constexpr int NT = L / 16, MT = (2 * DIN) / 16;        // 256, 24
  constexpr int NG = NT / 8;                             // 32 nt-groups
  __shared__ float tileA[16 * Dm];                       // 6 KB

  const int blk = blockIdx.x;
  const int ntg = blk % NG;
  const int mt  = (blk / NG) % MT;
  const int b   = blk / (NG * MT);
  const int m0  = mt * 16;

  // --- stage A tile: 16 rows x 96 cols = 1536 contiguous floats = 384 x 16B
  {
    const unsigned lbase = (unsigned)(size_t)&tileA[0];
    const float* wt = w + (size_t)m0 * Dm;
    for (int c = threadIdx.x; c < (16 * Dm) / 4; c += 256)
      async_g2l_b128(lbase + c * 16u, wt + c * 4);
    wait_asynccnt0();
    __syncthreads();
  }

  const int wave = threadIdx.x >> 5;
  const int lane = threadIdx.x & 31;
  const int nt   = ntg * 8 + wave;
  const int n0   = nt * 16;
  const int half = lane >> 4;                            // 0 | 1
  const int lid  = lane & 15;
  const float* xb = x + (size_t)b * Dm * L;

  v8f acc = {};
  for (int k0 = 0; k0 < Dm; k0 += 4) {
    const int ka = k0 + half * 2;                        // lanes 16-31: K+2,K+3
    v2f a, bt;
    a.x  = tileA[lid * Dm + ka];                         // ds_load_b64 (8B aligned)
    a.y  = tileA[lid * Dm + ka + 1];
    bt.x = xb[(size_t)ka * L + n0 + lid];
    bt.y = xb[(size_t)(ka + 1) * L + n0 + lid];
    acc = wmma_f32_16x16x4(a, bt, acc);
  }
  float* o = xz + ((size_t)b * (2 * DIN) + m0) * L + n0;
#pragma unroll
  for (int r = 0; r < 8; ++r)
    o[(size_t)(r + half * 8) * L + lid] = acc[r];
}

// ============================================================
// K2: depthwise 3x3 conv (SAME) + bias + SiLU  -> xs[b,d,l]
// ============================================================
__global__ void k_dwconv(const float* __restrict__ xz,
                         const float* __restrict__ cw,  // [192,1,3,3]
                         const float* __restrict__ cb,
                         float* __restrict__ xs) {
  const int i = blockIdx.x * blockDim.x + threadIdx.x;
  if (i >= Bsz * DIN * L) return;
  const int l = i % L;
  const int d = (i / L) % DIN;
  const int b = i / (L * DIN);
  const int h = l >> 6, w = l & 63;
  const float* src = xz + ((size_t)b * (2 * DIN) + d) * L;
  const float* k9  = cw + d * 9;
  float acc = cb[d];
#pragma unroll
  for (int dy = -1; dy <= 1; ++dy)
#pragma unroll
    for (int dx = -1; dx <= 1; ++dx) {
      const int hh = h + dy, ww = w + dx;
      if (hh >= 0 && hh < Himg && ww >= 0 && ww < Wimg)
        acc += k9[(dy + 1) * 3 + (dx + 1)] * src[(hh << 6) + ww];
    }
  xs[((size_t)b * DIN + d) * L + l] = silu_f(acc);
}

// ============================================================
// K3: x_proj  P[b,k,s,c] = sum_d xw[k,c,d] * xs[b,d,s]   (stored spatially)
//     layout P: [B, K, L, 8]  -> 8 contiguous floats per (b,k,s)
// ============================================================
__global__ void k_xproj(const float* __restrict__ xs,
                        const float* __restrict__ xw,   // [K,8,192]
                        float* __restrict__ P) {
  const int i = blockIdx.x * blockDim.x + threadIdx.x;
  if (i >= Bsz * L) return;
  const int s = i % L, b = i / L;
  float acc[KDIR * NCH];
#pragma unroll
  for (int j = 0; j < KDIR * NCH; ++j) acc[j] = 0.0f;
  const float* xb = xs + (size_t)b * DIN * L + s;
  for (int d = 0; d < DIN; ++d) {
    const float v = xb[(size_t)d * L];
#pragma unroll
    for (int j = 0; j < KDIR * NCH; ++j) acc[j] += xw[j * DIN + d] * v;
  }
#pragma unroll
  for (int k = 0; k < KDIR; ++k) {
    float* o = P + (((size_t)(b * KDIR + k)) * L + s) * NCH;
#pragma unroll
    for (int c = 0; c < NCH; ++c) o[c] = acc[k * NCH + c];
  }
}

// ============================================================
// K4: prior -> direction weights   WD[b,l,k] = 1 + sig(alpha)*(sig(mlp)-1)
// ============================================================
__global__ void k_prior(const float* __restrict__ prior, const float* __restrict__ alpha,
                        const float* __restrict__ w1, const float* __restrict__ b1,
                        const float* __restrict__ bnw, const float* __restrict__ bnb,
                        const float* __restrict__ bnm, const float* __restrict__ bnv,
                        const float* __restrict__ w2, const float* __restrict__ b2,
                        float* __restrict__ WD) {
  const int i = blockIdx.x * blockDim.x + threadIdx.x;
  if (i >= Bsz * L) return;
  const int l = i % L, b = i / L;
  float pr[4];
#pragma unroll
  for (int c = 0; c < 4; ++c) pr[c] = prior[((size_t)b * 4 + c) * L + l];
  const float sa = sigmoid_f(alpha[0]);
  float g[4] = {b2[0], b2[1], b2[2], b2[3]};
  for (int j = 0; j < HID; ++j) {
    float t = b1[j];
#pragma unroll
    for (int c = 0; c < 4; ++c) t += w1[j * 4 + c] * pr[c];
    t = (t - bnm[j]) * rsqrtf(bnv[j] + 1e-5f) * bnw[j] + bnb[j];
    t = fmaxf(t, 0.0f);
#pragma unroll
    for (int o = 0; o < 4; ++o) g[o] += w2[o * HID + j] * t;
  }
  float* o = WD + ((size_t)b * L + l) * 4;
#pragma unroll
  for (int k = 0; k < 4; ++k) o[k] = 1.0f + sa * (sigmoid_f(g[k]) - 1.0f);
}

// ------------- shared per-step scan math (N == 1) ---------------------------
__device__ __forceinline__ void ss_step(const float* __restrict__ P8, float xv,
                                        const float dtw[RLOW], float dtb, float Aval,
                                        float& a, float& bx) {
  float dtr = dtb;
#pragma unroll
  for (int r = 0; r < RLOW; ++r) dtr += dtw[r] * P8[r];
  const float dt = softplus_f(dtr);
  a  = __expf(dt * Aval);
  bx = dt * P8[6] * xv;
}

// ============================================================
// K5a: scan phase A — per-chunk affine summary (hA = prod a, hB)
// ============================================================
__global__ void k_scanA(const float* __restrict__ P, const float* __restrict__ xs,
                        const float* __restrict__ dtwg,  // [K,192,6]
                        const float* __restrict__ dtbg,  // [K,192]
                        const float* __restrict__ alog,  // [K*192, 1]
                        float* __restrict__ HA, float* __restrict__ HB) {
  const int i = blockIdx.x * blockDim.x + threadIdx.x;
  if (i >= Bsz * KDIR * DIN * NCHK) return;
  const int c = i % NCHK;
  const int d = (i / NCHK) % DIN;
  const int k = (i / (NCHK * DIN)) % KDIR;
  const int b = i / (NCHK * DIN * KDIR);
  float dtw[RLOW];
#pragma unroll
  for (int r = 0; r < RLOW; ++r) dtw[r] = dtwg[(k * DIN + d) * RLOW + r];
  const float dtb  = dtbg[k * DIN + d];
  const float Aval = -__expf(alog[k * DIN + d]);
  const float* Pk = P + ((size_t)(b * KDIR + k)) * L * NCH;
  const float* xb = xs + ((size_t)b * DIN + d) * L;
  float hA = 1.0f, hB = 0.0f;
  for (int l = c * CLEN; l < (c + 1) * CLEN; ++l) {
    const int s = dir_s(k, l);
    if ((l & 7) == 0) {                    // stride prefetch 8 steps ahead
      const int lp = (l + 8 < L) ? l + 8 : l;
      __builtin_prefetch(Pk + (size_t)dir_s(k, lp) * NCH, 0, 1);
    }
    const float* P8 = Pk + (size_t)s * NCH;
    float a, bx;
    ss_step(P8, xb[s], dtw, dtb, Aval, a, bx);
    hA *= a;
    hB = hB * a + bx;
  }
  HA[i] = hA; HB[i] = hB;
}

// K5b: per-chain prefix over the 8 chunk summaries -> h_in per chunk
__global__ void k_scanPrefix(const float* __restrict__ HA, const float* __restrict__ HB,
                             float* __restrict__ HIN) {
  const int i = blockIdx.x * blockDim.x + threadIdx.x;
  if (i >= Bsz * KDIR * DIN) return;
  const int base = i * NCHK;
  float h = 0.0f;
#pragma unroll
  for (int c = 0; c < NCHK; ++c) {
    HIN[base + c] = h;
    h = HA[base + c] * h + HB[base + c];
  }
}

// K5c: phase C — replay chunk from prefix state, emit weighted y (seq order)
__global__ void k_scanC(const float* __restrict__ P, const float* __restrict__ xs,
                        const float* __restrict__ dtwg, const float* __restrict__ dtbg,
                        const float* __restrict__ alog, const float* __restrict__ Ds,
                        const float* __restrict__ WD, const float* __restrict__ HIN,
                        float* __restrict__ YSW) {
  const int i = blockIdx.x * blockDim.x + threadIdx.x;
  if (i >= Bsz * KDIR * DIN * NCHK) return;
  const int c = i % NCHK;
  const int d = (i / NCHK) % DIN;
  const int k = (i / (NCHK * DIN)) % KDIR;
  const int b = i / (NCHK * DIN * KDIR);
  float dtw[RLOW];
#pragma unroll
  for (int r = 0; r < RLOW; ++r) dtw[r] = dtwg[(k * DIN + d) * RLOW + r];
  const float dtb  = dtbg[k * DIN + d];
  const float Aval = -__expf(alog[k * DIN + d]);
  const float Dv   = Ds[k * DIN + d];
  const float* Pk = P + ((size_t)(b * KDIR + k)) * L * NCH;
  const float* xb = xs + ((size_t)b * DIN + d) * L;
  const float* wd = WD + (size_t)b * L * 4 + k;
  float* yo = YSW + (((size_t)(b * KDIR + k)) * DIN + d) * L;
  float h = HIN[i];
  for (int l = c * CLEN; l < (c + 1) * CLEN; ++l) {
    const int s = dir_s(k, l);
    if ((l & 7) == 0) {
      const int lp = (l + 8 < L) ? l + 8 : l;
      __builtin_prefetch(Pk + (size_t)dir_s(k, lp) * NCH, 0, 1);
    }
    const float* P8 = Pk + (size_t)s * NCH;
    const float xv = xb[s];
    float a, bx;
    ss_step(P8, xv, dtw, dtb, Aval, a, bx);
    h = h * a + bx;
    float y = h * P8[7] + Dv * xv;          // y = h*C + D*x   (N == 1)
    y *= wd[(size_t)l * 4];                 // direction weight at seq index l
    yo[l] = y;
  }
}

// ============================================================
// K6: cross-merge (inverse permutations, sum of 4 dirs) -> ym[b,d,s]
// ============================================================
__global__ void k_merge(const float* __restrict__ YSW, float* __restrict__ YM) {
  const int i = blockIdx.x * blockDim.x + threadIdx.x;
  if (i >= Bsz * DIN * L) return;
  const int s = i % L;
  const int d = (i / L) % DIN;
  const int b = i / (L * DIN);
  const int lc = ((s & 63) << 6) | (s >> 6);          // col-major seq index
  const size_t row = (size_t)DIN * L;
  const float* y0 = YSW + (((size_t)(b * KDIR + 0)) * DIN + d) * L;
  const float* y1 = y0 + row;
  const float* y2 = y1 + row;
  const float* y3 = y2 + row;
  YM[((size_t)b * DIN + d) * L + s] =
      y0[s] + y2[L - 1 - s] + y1[lc] + y3[L - 1 - lc];
}

// ============================================================
// K7: channel LayerNorm + SiLU(z) gate -> yact[b,d,s]
// ============================================================
__global__ void k_ln_gate(const float* __restrict__ YM, const float* __restrict__ xz,
                          const float* __restrict__ onw, const float* __restrict__ onb,
                          float* __restrict__ YACT) {
  const int i = blockIdx.x * blockDim.x + threadIdx.x;
  if (i >= Bsz * L) return;
  const int s = i % L, b = i / L;
  const float* yb = YM + (size_t)b * DIN * L + s;
  float sum = 0.0f, sq = 0.0f;
  for (int d = 0; d < DIN; ++d) {
    const float v = yb[(size_t)d * L];
    sum += v; sq += v * v;
  }
  const float mu = sum * (1.0f / DIN);
  const float var = sq * (1.0f / DIN) - mu * mu;
  const float rstd = rsqrtf(var + 1e-5f);
  const float* zb = xz + ((size_t)b * (2 * DIN) + DIN) * L + s;
  float* ob = YACT + (size_t)b * DIN * L + s;
  for (int d = 0; d < DIN; ++d) {
    const float v  = (yb[(size_t)d * L] - mu) * rstd * onw[d] + onb[d];
    const float z  = zb[(size_t)d * L];
    ob[(size_t)d * L] = v * silu_f(z);
  }
}

// ============================================================
// K8: out_proj GEMM  out[b,m(96),l] = sum_d Wo[m,d] * yact[b,d,l]  (WMMA)
//     Same LDS-staged A-tile scheme as K1 (16x192 = 12KB contiguous).
// ============================================================
__global__ void k_out_proj(const float* __restrict__ yact,
                           const float* __restrict__ w,   // [96, 192]
                           float* __restrict__ out) {
  constexpr int NT = L / 16, MT = Dm / 16;               // 256, 6
  constexpr int NG = NT / 8;                             // 32
  __shared__ float tileA[16 * DIN];                      // 12 KB

  const int blk = blockIdx.x;
  const int ntg = blk % NG;
  const int mt  = (blk / NG) % MT;
  const int b   = blk / (NG * MT);
  const int m0  = mt * 16;

  {
    const unsigned lbase = (unsigned)(size_t)&tileA[0];
    const float* wt = w + (size_t)m0 * DIN;
    for (int c = threadIdx.x; c < (16 * DIN) / 4; c += 256)
      async_g2l_b128(lbase + c * 16u, wt + c * 4);
    wait_asynccnt0();
    __syncthreads();
  }

  const int wave = threadIdx.x >> 5;
  const int lane = threadIdx.x & 31;
  const int nt   = ntg * 8 + wave;
  const int n0   = nt * 16;
  const int half = lane >> 4;
  const int lid  = lane & 15;
  const float* yb = yact + (size_t)b * DIN * L;

  v8f acc = {};
  for (int k0 = 0; k0 < DIN; k0 += 4) {
    const int ka = k0 + half * 2;
    v2f a, bt;
    a.x  = tileA[lid * DIN + ka];
    a.y  = tileA[lid * DIN + ka + 1];
    bt.x = yb[(size_t)ka * L + n0 + lid];
    bt.y = yb[(size_t)(ka + 1) * L + n0 + lid];
    acc = wmma_f32_16x16x4(a, bt, acc);
  }
  float* o = out + ((size_t)b * Dm + m0) * L + n0;
#pragma unroll
  for (int r = 0; r < 8; ++r)
    o[(size_t)(r + half * 8) * L + lid] = acc[r];
}

// ============================================================
extern "C" void kernel_launch(void* const* d_in, const int* in_sizes, int n_in,
                              void* d_out, int out_size, void* d_ws, size_t ws_size,
                              hipStream_t stream) {
  const float* x      = (const float*)d_in[0];
  const float* prior  = (const float*)d_in[1];
  const float* alpha  = (const float*)d_in[2];
  const float* ipw    = (const float*)d_in[3];
  const float* cw     = (const float*)d_in[4];
  const float* cb     = (const float*)d_in[5];
  const float* xpw    = (const float*)d_in[6];
  const float* dtw    = (const float*)d_in[7];
  const float* dtb    = (const float*)d_in[8];
  const float* alog   = (const float*)d_in[9];
  const float* Ds     = (const float*)d_in[10];
  const float* onw    = (const float*)d_in[11];
  const float* onb    = (const float*)d_in[12];
  const float* opw    = (const float*)d_in[13];
  const float* w1     = (const float*)d_in[14];
  const float* b1     = (const float*)d_in[15];
  const float* bnw    = (const float*)d_in[16];
  const float* bnb    = (const float*)d_in[17];
  const float* bnm    = (const float*)d_in[18];
  const float* bnv    = (const float*)d_in[19];
  const float* w2     = (const float*)d_in[20];
  const float* b2     = (const float*)d_in[21];
  float* out = (float*)d_out;

  // ---- workspace layout (floats) ----
  float* XZ  = (float*)d_ws;                              // [B,384,L]
  float* XS  = XZ  + (size_t)Bsz * 2 * DIN * L;           // [B,192,L] (reused as YACT)
  float* P   = XS  + (size_t)Bsz * DIN * L;               // [B,K,L,8]
  float* WD  = P   + (size_t)Bsz * KDIR * L * NCH;        // [B,L,4]
  float* HA  = WD  + (size_t)Bsz * L * 4;                 // [B,K,192,NCHK]
  float* HB  = HA  + (size_t)Bsz * KDIR * DIN * NCHK;
  float* HIN = HB  + (size_t)Bsz * KDIR * DIN * NCHK;
  float* YSW = HIN + (size_t)Bsz * KDIR * DIN * NCHK;     // [B,K,192,L]
  float* YM  = YSW + (size_t)Bsz * KDIR * DIN * L;        // [B,192,L]
  float* YACT = XS;                                       // alias: XS dead after scan

  const int TPB = 256;
  // K1: in_proj WMMA GEMM (LDS-staged A tiles, 8 waves/block)
  {
    const int blocks = Bsz * ((2 * DIN) / 16) * ((L / 16) / 8);
    k_in_proj<<<blocks, TPB, 0, stream>>>(x, ipw, XZ);
  }
  // K2: depthwise conv + SiLU
  {
    const int n = Bsz * DIN * L;
    k_dwconv<<<(n + TPB - 1) / TPB, TPB, 0, stream>>>(XZ, cw, cb, XS);
  }
  // K3: x_proj (8 low-rank channels per direction)
  {
    const int n = Bsz * L;
    k_xproj<<<(n + TPB - 1) / TPB, TPB, 0, stream>>>(XS, xpw, P);
  }
  // K4: prior MLP -> direction weights
  {
    const int n = Bsz * L;
    k_prior<<<(n + TPB - 1) / TPB, TPB, 0, stream>>>(prior, alpha, w1, b1, bnw,
                                                     bnb, bnm, bnv, w2, b2, WD);
  }
  // K5: chunked selective scan (phase A, prefix, phase C)
  {
    const int nA = Bsz * KDIR * DIN * NCHK;
    k_scanA<<<(nA + TPB - 1) / TPB, TPB, 0, stream>>>(P, XS, dtw, dtb, alog, HA, HB);
    const int nP = Bsz * KDIR * DIN;
    k_scanPrefix<<<(nP + TPB - 1) / TPB, TPB, 0, stream>>>(HA, HB, HIN);
    k_scanC<<<(nA + TPB - 1) / TPB, TPB, 0, stream>>>(P, XS, dtw, dtb, alog, Ds,
                                                      WD, HIN, YSW);
  }
  // K6: cross-merge
  {
    const int n = Bsz * DIN * L;
    k_merge<<<(n + TPB - 1) / TPB, TPB, 0, stream>>>(YSW, YM);
  }
  // K7: LayerNorm (channel) + SiLU gate
  {
    const int n = Bsz * L;
    k_ln_gate<<<(n + TPB - 1) / TPB, TPB, 0, stream>>>(YM, XZ, onw, onb, YACT);
  }
  // K8: out_proj WMMA GEMM -> d_out
  {
    const int blocks = Bsz * (Dm / 16) * ((L / 16) / 8);
    k_out_proj<<<blocks, TPB, 0, stream>>>(YACT, opw, out);
  }
}